// Classifier_7275674600362
// MI455X (gfx1250) — compile-verified
//
#include <hip/hip_runtime.h>
#include <math.h>

// Problem dims
#define S_SEG 256
#define T_TOK 512
#define BERTD 768
#define POSD  128
#define HID   1024
#define NCLS  6
#define NCOM  64

typedef __attribute__((ext_vector_type(2))) float v2f;
typedef __attribute__((ext_vector_type(8))) float v8f;

// ---------------------------------------------------------------------------
// Kernel 1: per-segment attention logits + softmax + weighted pooling.
// One block (256 thr = 8 waves) per segment. emb is streamed once from HBM;
// the pooling pass re-reads the same 1.5MB tile which stays resident in L2.
// ---------------------------------------------------------------------------
__global__ __launch_bounds__(256) void attn_pool_kernel(
    const float* __restrict__ emb, const float* __restrict__ pos,
    const float* __restrict__ Wa, const float* __restrict__ ba,
    float* __restrict__ segvec)
{
    __shared__ float s_wa[BERTD + POSD];   // 896 attention weights
    __shared__ float s_w[T_TOK];           // logits -> softmax weights
    __shared__ float s_red[256];

    const int tid  = threadIdx.x;
    const int s    = blockIdx.x;
    const int lane = tid & 31;
    const int wave = tid >> 5;

    for (int i = tid; i < BERTD + POSD; i += 256) s_wa[i] = Wa[i];
    __syncthreads();

    const float battn = ba[0];
    const float4* wa_e4 = (const float4*)s_wa;             // 192 x float4
    const float4* wa_p4 = (const float4*)(s_wa + BERTD);   //  32 x float4

    // ---- pass 1: logits, one token per wave iteration (coalesced float4) ----
    for (int t = wave; t < T_TOK; t += 8) {
        const float4* e4 = (const float4*)(emb + ((size_t)s * T_TOK + t) * BERTD);
        const float4* p4 = (const float4*)(pos + ((size_t)s * T_TOK + t) * POSD);
        float acc = 0.f;
        #pragma unroll
        for (int i = 0; i < 6; ++i) {
            float4 e = e4[lane + 32 * i];
            float4 w = wa_e4[lane + 32 * i];
            acc += e.x * w.x + e.y * w.y + e.z * w.z + e.w * w.w;
        }
        {
            float4 p = p4[lane];
            float4 w = wa_p4[lane];
            acc += p.x * w.x + p.y * w.y + p.z * w.z + p.w * w.w;
        }
        #pragma unroll
        for (int off = 16; off > 0; off >>= 1)       // wave32 reduction
            acc += __shfl_xor(acc, off, 32);
        if (lane == 0) s_w[t] = acc + battn;
    }
    __syncthreads();

    // ---- softmax over T=512 ----
    float lmax = -INFINITY;
    for (int t = tid; t < T_TOK; t += 256) lmax = fmaxf(lmax, s_w[t]);
    s_red[tid] = lmax;
    __syncthreads();
    for (int o = 128; o > 0; o >>= 1) {
        if (tid < o) s_red[tid] = fmaxf(s_red[tid], s_red[tid + o]);
        __syncthreads();
    }
    const float m = s_red[0];
    __syncthreads();

    float lsum = 0.f;
    for (int t = tid; t < T_TOK; t += 256) {
        float e = expf(s_w[t] - m);
        s_w[t] = e;
        lsum += e;
    }
    s_red[tid] = lsum;
    __syncthreads();
    for (int o = 128; o > 0; o >>= 1) {
        if (tid < o) s_red[tid] += s_red[tid + o];
        __syncthreads();
    }
    const float inv = 1.f / s_red[0];
    __syncthreads();
    for (int t = tid; t < T_TOK; t += 256) s_w[t] *= inv;
    __syncthreads();

    // ---- pass 2: weighted pooling (768 cols across 256 threads, coalesced) ----
    float a0 = 0.f, a1 = 0.f, a2 = 0.f;
    const float* ebase = emb + (size_t)s * T_TOK * BERTD;
    for (int t = 0; t < T_TOK; ++t) {
        const float w = s_w[t];
        const float* row = ebase + (size_t)t * BERTD;
        // stream-ahead hint (lowers to global_prefetch_b8; speculative, OOB ok)
        __builtin_prefetch(row + 8 * BERTD + tid, 0, 3);
        a0 += w * row[tid];
        a1 += w * row[tid + 256];
        a2 += w * row[tid + 512];
    }
    float* o = segvec + (size_t)s * BERTD;
    o[tid] = a0; o[tid + 256] = a1; o[tid + 512] = a2;
}

// ---------------------------------------------------------------------------
// Kernel 2: deterministic ragged segment-sum (segment_ids sorted, 64 comments).
// One block per comment scans the 256 ids; each segvec row is read once.
// ---------------------------------------------------------------------------
__global__ __launch_bounds__(256) void segsum_kernel(
    const float* __restrict__ segvec, const int* __restrict__ segid,
    float* __restrict__ vecs)
{
    const int c   = blockIdx.x;
    const int tid = threadIdx.x;
    float a0 = 0.f, a1 = 0.f, a2 = 0.f;
    for (int s = 0; s < S_SEG; ++s) {
        if (segid[s] == c) {
            const float* r = segvec + (size_t)s * BERTD;
            a0 += r[tid]; a1 += r[tid + 256]; a2 += r[tid + 512];
        }
    }
    float* o = vecs + (size_t)c * BERTD;
    o[tid] = a0; o[tid + 256] = a1; o[tid + 512] = a2;
}

// ---------------------------------------------------------------------------
// Kernel 3: fp32 WMMA GEMM  C(64xN) = LeakyReLU(A(64xK) @ B(KxN) + bias)
// via V_WMMA_F32_16X16X4_F32. K, N compile-time -> zero bounds checks, no
// exec-mask juggling. 4 waves per block, each wave owns one 16-row tile
// (M=64 exactly); one block per 16-column tile.
//
// B is staged through LDS once per block (shared by all 4 waves). LDS layout
// packs the two K-values a lane needs adjacently:  s_b[pair][n][2], so the
// B fetch is a single ds_load_b64, and lane-halves 0/1 read even/odd row
// pairs -> disjoint 32-bank halves (conflict-free).
//
// ISA f32 operand layouts (ISA 7.12.2):
//   A 16x4 : M = lane&15, lane-half selects K pair, regs = K within pair
//   B 4x16 : N = lane&15, lane-half selects K pair (mirrored)
//   C/D    : VGPR r -> (M = r + 8*(lane>>4), N = lane&15)
// ---------------------------------------------------------------------------
template <int K, int N>
__global__ __launch_bounds__(128) void gemm64_wmma_kernel(
    const float* __restrict__ A, const float* __restrict__ B,
    const float* __restrict__ bias, float* __restrict__ C)
{
    constexpr int KB = 32;                 // K-chunk staged per block
    __shared__ float s_b[KB * 16];         // [pair][n][2] packing, 2KB

    const int tid  = threadIdx.x;
    const int lane = tid & 31;
    const int wave = tid >> 5;
    const int half = lane >> 4;            // 0: lanes 0-15, 1: lanes 16-31
    const int mt   = lane & 15;
    const int nt   = lane & 15;
    const int kb   = half * 2;             // K offset pair within a K=4 step

    const int m0   = wave * 16;
    const int n0   = blockIdx.x * 16;

    const int lrow = tid >> 4;             // staging: 8 rows per pass
    const int lcol = tid & 15;

    v8f acc = {0.f, 0.f, 0.f, 0.f, 0.f, 0.f, 0.f, 0.f};
    const float* arow = A + (size_t)(m0 + mt) * K;

    for (int k0 = 0; k0 < K; k0 += KB) {
        __syncthreads();
        #pragma unroll
        for (int i = 0; i < KB / 8; ++i) {
            const int row = lrow + i * 8;
            // pack: element (row, col) -> s_b[(row>>1)*32 + col*2 + (row&1)]
            s_b[(row >> 1) * 32 + lcol * 2 + (row & 1)] =
                B[(size_t)(k0 + row) * N + n0 + lcol];
        }
        __syncthreads();

        #pragma unroll
        for (int kk = 0; kk < KB; kk += 4) {
            v2f a, b;
            a.x = arow[k0 + kk + kb];          // contiguous -> global_load_b64
            a.y = arow[k0 + kk + kb + 1];
            const float2 bp = *(const float2*)&s_b[((kk + kb) >> 1) * 32 + nt * 2];
            b.x = bp.x;                        // one ds_load_b64
            b.y = bp.y;
            acc = __builtin_amdgcn_wmma_f32_16x16x4_f32(
                /*neg_a=*/false, a, /*neg_b=*/false, b,
                /*c_mod=*/(short)0, acc, /*reuse_a=*/false, /*reuse_b=*/false);
        }
    }

    const float bv = bias[n0 + nt];
    #pragma unroll
    for (int r = 0; r < 8; ++r) {
        const int row = m0 + r + half * 8;
        float v = acc[r] + bv;
        v = (v > 0.f) ? v : 0.01f * v;         // LeakyReLU(0.01)
        C[(size_t)row * N + n0 + nt] = v;
    }
}

// ---------------------------------------------------------------------------
// Kernel 4: final 64x1024 @ 1024x6 + sigmoid. 384 dot products -- noise.
// One block per class, one thread per comment; fully deterministic.
// ---------------------------------------------------------------------------
__global__ __launch_bounds__(64) void head_kernel(
    const float* __restrict__ H, const float* __restrict__ W3,
    const float* __restrict__ b3, float* __restrict__ out)
{
    const int c = threadIdx.x;      // comment 0..63
    const int j = blockIdx.x;       // class 0..5
    const float* h = H + (size_t)c * HID;
    float acc = 0.f;
    for (int k = 0; k < HID; ++k) acc += h[k] * W3[k * NCLS + j];
    out[c * NCLS + j] = 1.f / (1.f + expf(-(acc + b3[j])));
}

// ---------------------------------------------------------------------------
extern "C" void kernel_launch(void* const* d_in, const int* in_sizes, int n_in,
                              void* d_out, int out_size, void* d_ws, size_t ws_size,
                              hipStream_t stream) {
    const float* emb = (const float*)d_in[0];
    const float* pos = (const float*)d_in[1];
    const float* Wa  = (const float*)d_in[2];
    const float* ba  = (const float*)d_in[3];
    const float* W1  = (const float*)d_in[4];
    const float* b1  = (const float*)d_in[5];
    const float* W2  = (const float*)d_in[6];
    const float* b2  = (const float*)d_in[7];
    const float* W3  = (const float*)d_in[8];
    const float* b3  = (const float*)d_in[9];
    const int* segid = (const int*)d_in[10];
    float* out = (float*)d_out;

    float* ws     = (float*)d_ws;
    float* segvec = ws;                          // 256*768
    float* vecs   = segvec + S_SEG * BERTD;      // 64*768
    float* h1     = vecs + NCOM * BERTD;         // 64*1024
    float* h2     = h1 + NCOM * HID;             // 64*1024

    attn_pool_kernel<<<S_SEG, 256, 0, stream>>>(emb, pos, Wa, ba, segvec);
    segsum_kernel<<<NCOM, 256, 0, stream>>>(segvec, segid, vecs);
    gemm64_wmma_kernel<BERTD, HID><<<HID / 16, 128, 0, stream>>>(vecs, W1, b1, h1);
    gemm64_wmma_kernel<HID,   HID><<<HID / 16, 128, 0, stream>>>(h1,   W2, b2, h2);
    head_kernel<<<NCLS, NCOM, 0, stream>>>(h2, W3, b3, out);
}